// Median_58849641890432
// MI455X (gfx1250) — compile-verified
//
#include <hip/hip_runtime.h>

// Median-of-49 per pixel, [Y,X,L] f32 -> [Y,X] f32 (lower median, L odd -> exact).
//
// MI455X plan:
//  * 392 MB read-once / 8 MB write => ~17us HBM floor @ 23.3 TB/s. Read path
//    uses gfx1250 async global->LDS copies with th:TH_LOAD_NT (stream past the
//    192MB L2), tracked by ASYNCcnt and waited with s_wait_asynccnt.
//  * Staging is WAVE-LOCAL: wave w of a block owns pixels 32w..32w+31, whose
//    32*49 floats are the contiguous 6272B at LDS offset w*6272. Each wave
//    copies its own chunk and waits only on its own ASYNCcnt -- no barrier.
//  * Median via forgetful selection: 26-float register window, compare-exchange
//    network pushes min/max out, refill, shrink 26->4, v_med3_f32 of the 3
//    survivors. All indices compile-time => window lives in VGPRs.

#define LVOL 49
#define PPB  256   // threads per block (8 waves), 1 pixel/thread
#define HALF 26    // forgetful-selection initial window: ceil(49/2)+1

__device__ __forceinline__ void cas(float& a, float& b) {
    float lo = fminf(a, b);
    float hi = fmaxf(a, b);
    a = lo; b = hi;
}

// After mnmx<W>: v[0] = min of v[0..W-1], v[W-1] = max of v[0..W-1].
template<int W>
__device__ __forceinline__ void mnmx(float* v) {
    constexpr int OFF = (W + 1) / 2;
    constexpr int H   = W / 2;
    #pragma unroll
    for (int i = 0; i < W - OFF; ++i) cas(v[i], v[i + OFF]);   // pair halves
    #pragma unroll
    for (int i = 1; i <= H; ++i)      cas(v[0], v[i]);         // min chain
    #pragma unroll
    for (int i = H; i <= W - 2; ++i)  cas(v[i], v[W - 1]);     // max chain
}

// Window v[0..W-1]: drop min (overwrite v[0] with next element) and max
// (v[W-1] falls out of the next, smaller window). Recurse W=26..4.
template<int W>
struct Forget {
    static __device__ __forceinline__ void run(float* v, const float* x, int& idx) {
        mnmx<W>(v);
        v[0] = x[idx]; ++idx;
        Forget<W - 1>::run(v, x, idx);
    }
};
template<>
struct Forget<3> {
    static __device__ __forceinline__ void run(float*, const float*, int&) {}
};

// Per wave: 12 b128 rounds (32 lanes * 16B * 12 = 6144 B) + 32*4B tail = 6272 B.
#define ASYNC_B128(OFS)                                                         \
    asm volatile("global_load_async_to_lds_b128 %0, %1, %2 offset:" #OFS        \
                 " th:TH_LOAD_NT"                                               \
                 :: "v"(vdst128), "v"(voff128), "s"(srcp) : "memory")

__global__ __launch_bounds__(PPB) void median49_kernel(
        const float* __restrict__ vol, float* __restrict__ out) {
    __shared__ float smem[PPB * LVOL];            // 50176 B

    const int t    = threadIdx.x;
    const int lane = t & 31;
    const int wave = t >> 5;

    // Wave-local source chunk: block base + wave * 32*49 floats (6272 B).
    const size_t base = (size_t)blockIdx.x * (PPB * LVOL) + (size_t)wave * (32 * LVOL);
    const float* srcp = vol + base;               // uniform per wave -> SGPR pair

    unsigned lds_base = (unsigned)(unsigned long long)(&smem[0]);
    unsigned ldsw     = lds_base + (unsigned)wave * 6272u;
    unsigned voff128  = (unsigned)lane * 16u;
    unsigned vdst128  = ldsw + voff128;

    ASYNC_B128(0);    ASYNC_B128(512);  ASYNC_B128(1024); ASYNC_B128(1536);
    ASYNC_B128(2048); ASYNC_B128(2560); ASYNC_B128(3072); ASYNC_B128(3584);
    ASYNC_B128(4096); ASYNC_B128(4608); ASYNC_B128(5120); ASYNC_B128(5632);

    unsigned voff32 = (unsigned)lane * 4u;
    unsigned vdst32 = ldsw + voff32;
    asm volatile("global_load_async_to_lds_b32 %0, %1, %2 offset:6144 th:TH_LOAD_NT"
                 :: "v"(vdst32), "v"(voff32), "s"(srcp) : "memory");

    // Wave-local completion: ASYNCcnt is per-wave, and this wave only reads
    // the LDS region it wrote itself -- no workgroup barrier needed.
    asm volatile("s_wait_asynccnt 0" ::: "memory");

    // ---- per-pixel forgetful selection out of LDS ----
    const float* x = &smem[t * LVOL];   // stride 49 dwords: bank-conflict-free
    float v[HALF];
    #pragma unroll
    for (int i = 0; i < HALF; ++i) v[i] = x[i];
    int idx = HALF;
    Forget<HALF>::run(v, x, idx);       // consumes x[26..48], leaves v[0..2]

    // median of the 3 survivors == global lower median (single v_med3_f32)
    float med = __builtin_amdgcn_fmed3f(v[0], v[1], v[2]);

    out[(size_t)blockIdx.x * PPB + t] = med;
}

extern "C" void kernel_launch(void* const* d_in, const int* in_sizes, int n_in,
                              void* d_out, int out_size, void* d_ws, size_t ws_size,
                              hipStream_t stream) {
    const float* vol = (const float*)d_in[0];
    float*       out = (float*)d_out;
    const int npix    = in_sizes[0] / LVOL;   // 1024*2048 = 2,097,152
    const int nblocks = npix / PPB;           // 8192 (exact)
    median49_kernel<<<dim3(nblocks), dim3(PPB), 0, stream>>>(vol, out);
}